// DeepFRI_51307679318435
// MI455X (gfx1250) — compile-verified
//
#include <hip/hip_runtime.h>
#include <hip/hip_bf16.h>
#include <math.h>

#define LNODES 8192
#define NEDGES 524288
#define NEG_SLOPE 0.2f
#define KPAD 32

typedef __attribute__((ext_vector_type(2))) float v2f;
typedef __attribute__((ext_vector_type(8))) float v8f;

// ---------------------------------------------------------------------------
// Padding helpers (run once, tiny): widen seq K 26->32, widen W1 rows 26->32.
// ---------------------------------------------------------------------------
__global__ void pad_cols(const float* __restrict__ src, float* __restrict__ dst,
                         int rows, int scols, int dcols)
{
    int idx = blockIdx.x * 256 + threadIdx.x;
    if (idx >= rows * dcols) return;
    int r = idx / dcols, c = idx - r * dcols;
    dst[idx] = (c < scols) ? src[(size_t)r * scols + c] : 0.f;
}
__global__ void pad_rows(const float* __restrict__ src, float* __restrict__ dst,
                         int srows, int drows, int cols)
{
    int idx = blockIdx.x * 256 + threadIdx.x;
    if (idx >= drows * cols) return;
    dst[idx] = (idx < srows * cols) ? src[idx] : 0.f;
}

// ---------------------------------------------------------------------------
// GEMM C[M,N] = A[M,K] @ B[K,N] (row-major), K % 4 == 0, K >= 8, optional ReLU.
// V_WMMA_F32_16X16X4_F32; one wave owns a 16x64 strip (4 tiles share A frag).
// Software-pipelined: next k-step's fragments are loaded as a clause before
// the current step's 4 WMMAs issue -> partial s_wait_loadcnt, overlap.
// ---------------------------------------------------------------------------
__global__ void __launch_bounds__(128)
gemm_f32_wmma(const float* __restrict__ A, const float* __restrict__ B,
              float* __restrict__ C, int N, int K, int relu)
{
    const int lane = threadIdx.x & 31;
    const int wave = threadIdx.x >> 5;
    const int half = lane >> 4;   // 0: lanes 0-15, 1: lanes 16-31
    const int r    = lane & 15;
    const int m0 = (blockIdx.y * 4 + wave) * 16;
    const int n0 = blockIdx.x * 64;

    // A fragment: lane holds K = k+2*half, k+2*half+1 of row m0+r (8B aligned)
    const float* ap = A + (size_t)(m0 + r) * K + 2 * half;
    // B fragment: b.x from row k+2*half, b.y from row k+2*half+1, col n0+16t+r
    const float* bp = B + (size_t)(2 * half) * N + n0 + r;
    const size_t bstep = (size_t)4 * N;

    v8f acc[4] = {v8f{}, v8f{}, v8f{}, v8f{}};

    // prologue: load first fragments
    v2f a0 = *(const v2f*)ap;
    ap += 4;
    v2f bf[4];
#pragma unroll
    for (int t = 0; t < 4; ++t) {
        bf[t].x = bp[t * 16];
        bf[t].y = bp[N + t * 16];
    }
    bp += bstep;

#pragma unroll 2
    for (int k = 4; k < K; k += 4) {
        // issue next iteration's loads (clause) before current WMMAs
        v2f an = *(const v2f*)ap;
        ap += 4;
        v2f bn[4];
#pragma unroll
        for (int t = 0; t < 4; ++t) {
            bn[t].x = bp[t * 16];
            bn[t].y = bp[N + t * 16];
        }
        bp += bstep;
#pragma unroll
        for (int t = 0; t < 4; ++t) {
            acc[t] = __builtin_amdgcn_wmma_f32_16x16x4_f32(
                false, a0, false, bf[t], (short)0, acc[t], false, false);
        }
        a0 = an;
#pragma unroll
        for (int t = 0; t < 4; ++t) bf[t] = bn[t];
    }
    // epilogue: last k-step
#pragma unroll
    for (int t = 0; t < 4; ++t) {
        acc[t] = __builtin_amdgcn_wmma_f32_16x16x4_f32(
            false, a0, false, bf[t], (short)0, acc[t], false, false);
    }

    // C/D layout: VGPR i -> M = m0 + 8*half + i, N = n0 + 16*t + r
#pragma unroll
    for (int t = 0; t < 4; ++t) {
#pragma unroll
        for (int i = 0; i < 8; ++i) {
            float v = acc[t][i];
            if (relu) v = fmaxf(v, 0.0f);
            C[(size_t)(m0 + 8 * half + i) * N + n0 + t * 16 + r] = v;
        }
    }
}

// ---------------------------------------------------------------------------
// Per-(node,head) attention projections: a_s[n,h] = <h[n,h,:], asrc[h,:]>
// ---------------------------------------------------------------------------
__global__ void attn_scores(const float* __restrict__ hmat,
                            const float* __restrict__ asrc,
                            const float* __restrict__ adst,
                            float* __restrict__ a_s, float* __restrict__ a_d,
                            int C)
{
    int idx = blockIdx.x * blockDim.x + threadIdx.x;
    if (idx >= LNODES * 4) return;
    int n = idx >> 2, h = idx & 3;
    const float* row = hmat + (size_t)n * (4 * C) + h * C;
    float ss = 0.f, sd = 0.f;
    for (int c = 0; c < C; ++c) {
        float v = row[c];
        ss += v * asrc[h * C + c];
        sd += v * adst[h * C + c];
    }
    a_s[idx] = ss;
    a_d[idx] = sd;
}

// ---------------------------------------------------------------------------
// CSR-by-destination build
// ---------------------------------------------------------------------------
__global__ void csr_count_init(int* counts)
{
    int n = blockIdx.x * 256 + threadIdx.x;
    if (n < LNODES) counts[n] = 1;   // self-loop
}
__global__ void csr_count_edges(const int* __restrict__ ei, int* counts)
{
    int e = blockIdx.x * 256 + threadIdx.x;
    if (e < NEDGES) atomicAdd(&counts[ei[NEDGES + e]], 1);
}
__global__ void __launch_bounds__(1024)
csr_scan(const int* __restrict__ counts, int* __restrict__ row_ptr)
{
    __shared__ int s[LNODES];
    const int tid = threadIdx.x;
#pragma unroll
    for (int k = 0; k < LNODES / 1024; ++k) s[tid + k * 1024] = counts[tid + k * 1024];
    __syncthreads();
    for (int off = 1; off < LNODES; off <<= 1) {
        int v[LNODES / 1024];
#pragma unroll
        for (int k = 0; k < LNODES / 1024; ++k) {
            int i = tid + k * 1024;
            v[k] = (i >= off) ? s[i - off] : 0;
        }
        __syncthreads();
#pragma unroll
        for (int k = 0; k < LNODES / 1024; ++k) s[tid + k * 1024] += v[k];
        __syncthreads();
    }
#pragma unroll
    for (int k = 0; k < LNODES / 1024; ++k) {
        int i = tid + k * 1024;
        row_ptr[i + 1] = s[i];
    }
    if (tid == 0) row_ptr[0] = 0;
}
__global__ void csr_selfloop(const int* __restrict__ row_ptr, int* cursor, int* col)
{
    int n = blockIdx.x * 256 + threadIdx.x;
    if (n < LNODES) {
        int p = row_ptr[n];
        col[p] = n;
        cursor[n] = p + 1;
    }
}
__global__ void csr_fill(const int* __restrict__ ei, int* cursor, int* col)
{
    int e = blockIdx.x * 256 + threadIdx.x;
    if (e < NEDGES) {
        int d = ei[NEDGES + e];
        int p = atomicAdd(&cursor[d], 1);
        col[p] = ei[e];
    }
}

// ---------------------------------------------------------------------------
// GAT aggregation: one 256-thread block per destination node.
// Local per-head max -> exp weights -> denom + weighted gather, no atomics.
// CT=512 uses float2 gathers (one global_load_b64 per edge-row per thread).
// ---------------------------------------------------------------------------
template <int CT>
__global__ void __launch_bounds__(256)
gat_aggregate(const float* __restrict__ hmat,
              const float* __restrict__ a_s,
              const float* __restrict__ a_d,
              const int* __restrict__ row_ptr,
              const int* __restrict__ col,
              const float* __restrict__ bias,
              float* __restrict__ out)
{
    constexpr int C   = CT / 4;    // channels per head
    constexpr int VEC = CT / 256;  // channels per thread (1 or 2, contiguous)
    const int n   = blockIdx.x;
    const int tid = threadIdx.x;
    const int start = row_ptr[n];
    const int deg   = row_ptr[n + 1] - start;

    __shared__ float s_red[4 * 256];
    __shared__ float s_w[256 * 4];
    __shared__ int   s_src[256];
    __shared__ float s_fin[8];   // [0..3]=max, [4..7]=denom

    float adn[4];
#pragma unroll
    for (int h = 0; h < 4; ++h) adn[h] = a_d[n * 4 + h];

    // pass 1: per-head max over incoming edges
    float mx[4] = {-3.0e38f, -3.0e38f, -3.0e38f, -3.0e38f};
    for (int i = tid; i < deg; i += 256) {
        int s = col[start + i];
#pragma unroll
        for (int h = 0; h < 4; ++h) {
            float e = a_s[s * 4 + h] + adn[h];
            e = (e > 0.f) ? e : NEG_SLOPE * e;
            mx[h] = fmaxf(mx[h], e);
        }
    }
#pragma unroll
    for (int h = 0; h < 4; ++h) s_red[h * 256 + tid] = mx[h];
    __syncthreads();
    for (int off = 128; off > 0; off >>= 1) {
        if (tid < off) {
#pragma unroll
            for (int h = 0; h < 4; ++h)
                s_red[h * 256 + tid] = fmaxf(s_red[h * 256 + tid], s_red[h * 256 + tid + off]);
        }
        __syncthreads();
    }
    if (tid < 4) s_fin[tid] = s_red[tid * 256];
    __syncthreads();
    float m[4];
#pragma unroll
    for (int h = 0; h < 4; ++h) m[h] = s_fin[h];

    const int c0   = VEC * tid;        // first owned channel
    const int hown = c0 / C;           // head of owned channels (VEC never straddles)
    float den[4] = {0.f, 0.f, 0.f, 0.f};
    float acc[VEC];
#pragma unroll
    for (int v = 0; v < VEC; ++v) acc[v] = 0.f;

    // pass 2: chunked exp weights + weighted gather of h[src] rows
    for (int base = 0; base < deg; base += 256) {
        __syncthreads();
        int i = base + tid;
        if (i < deg) {
            int s = col[start + i];
            s_src[tid] = s;
#pragma unroll
            for (int h = 0; h < 4; ++h) {
                float e = a_s[s * 4 + h] + adn[h];
                e = (e > 0.f) ? e : NEG_SLOPE * e;
                float wgt = __expf(e - m[h]);
                s_w[tid * 4 + h] = wgt;
                den[h] += wgt;
            }
        }
        __syncthreads();
        int cnt = min(256, deg - base);
        for (int j = 0; j < cnt; ++j) {
            const float* row = hmat + (size_t)s_src[j] * CT;
            float wgt = s_w[j * 4 + hown];
            if constexpr (VEC == 2) {
                v2f rv = *(const v2f*)(row + c0);   // global_load_b64
                acc[0] = fmaf(wgt, rv.x, acc[0]);
                acc[1] = fmaf(wgt, rv.y, acc[1]);
            } else {
                acc[0] = fmaf(wgt, row[c0], acc[0]);
            }
        }
    }
    __syncthreads();
#pragma unroll
    for (int h = 0; h < 4; ++h) s_red[h * 256 + tid] = den[h];
    __syncthreads();
    for (int off = 128; off > 0; off >>= 1) {
        if (tid < off) {
#pragma unroll
            for (int h = 0; h < 4; ++h)
                s_red[h * 256 + tid] += s_red[h * 256 + tid + off];
        }
        __syncthreads();
    }
    if (tid < 4) s_fin[4 + tid] = s_red[tid * 256];
    __syncthreads();
    float dinv = 1.0f / (s_fin[4 + hown] + 1e-16f);
#pragma unroll
    for (int v = 0; v < VEC; ++v) {
        int c = c0 + v;
        float o = acc[v] * dinv + bias[c];
        out[(size_t)n * CT + c] = fmaxf(o, 0.f);
    }
}

// ---------------------------------------------------------------------------
// Tail: column-sum pool, tiny FCs
// ---------------------------------------------------------------------------
__global__ void zero_f32(float* p, int n)
{
    int i = blockIdx.x * 256 + threadIdx.x;
    if (i < n) p[i] = 0.f;
}
__global__ void pool_cols512(const float* __restrict__ x, float* __restrict__ pooled)
{
    int tid = threadIdx.x;
    int r0 = blockIdx.x * 128;
    float a0 = 0.f, a1 = 0.f;
    for (int r = 0; r < 128; ++r) {
        const float* row = x + (size_t)(r0 + r) * 512;
        a0 += row[tid];
        a1 += row[tid + 256];
    }
    atomicAdd(pooled + tid, a0);
    atomicAdd(pooled + tid + 256, a1);
}
__global__ void fc_kernel(const float* __restrict__ x, const float* __restrict__ W,
                          const float* __restrict__ b, float* __restrict__ y,
                          int K, int N, int relu)
{
    int j = blockIdx.x * 256 + threadIdx.x;
    if (j >= N) return;
    float s = b[j];
    for (int k = 0; k < K; ++k) s = fmaf(x[k], W[(size_t)k * N + j], s);
    y[j] = relu ? fmaxf(s, 0.f) : s;
}

// ---------------------------------------------------------------------------
extern "C" void kernel_launch(void* const* d_in, const int* in_sizes, int n_in,
                              void* d_out, int out_size, void* d_ws, size_t ws_size,
                              hipStream_t stream)
{
    (void)in_sizes; (void)n_in; (void)out_size; (void)ws_size;
    const float* seq   = (const float*)d_in[0];   // [8192,26]
    const int*   ei    = (const int*)  d_in[1];   // [2,E]
    const float* W1    = (const float*)d_in[2];   // [26,1024]
    const float* W2    = (const float*)d_in[3];   // [1024,256]
    const float* Wg1   = (const float*)d_in[4];   // [256,256]
    const float* asrc1 = (const float*)d_in[5];   // [4,64]
    const float* adst1 = (const float*)d_in[6];
    const float* b1    = (const float*)d_in[7];   // [256]
    const float* Wg2   = (const float*)d_in[8];   // [256,512]
    const float* asrc2 = (const float*)d_in[9];   // [4,128]
    const float* adst2 = (const float*)d_in[10];
    const float* b2    = (const float*)d_in[11];  // [512]
    const float* fcW   = (const float*)d_in[12];  // [512,512]
    const float* fcb   = (const float*)d_in[13];
    const float* outW  = (const float*)d_in[14];  // [512,489]
    const float* outb  = (const float*)d_in[15];

    // workspace carve-out (regions reused across pipeline stages)
    char* w = (char*)d_ws;
    size_t cur = 0;
    auto alloc = [&](size_t bytes) -> void* {
        void* p = w + cur;
        cur = (cur + bytes + 255) & ~(size_t)255;
        return p;
    };
    float* reg0 = (float*)alloc((size_t)LNODES * 1024 * 4); // 33.5 MB: h0 | h1 | h2+out2
    float* h0   = reg0;                                      // [8192,1024]
    float* h1   = reg0;                                      // [8192,256]
    float* h2   = reg0;                                      // [8192,512]
    float* out2 = reg0 + (size_t)LNODES * 512;               // [8192,512]
    float* xbuf = (float*)alloc((size_t)LNODES * 256 * 4);   // x -> out1
    float* out1 = xbuf;
    float* seqp = (float*)alloc((size_t)LNODES * KPAD * 4);  // padded seq [8192,32]
    float* W1p  = (float*)alloc((size_t)KPAD * 1024 * 4);    // padded W1 [32,1024]
    float* as1 = (float*)alloc((size_t)LNODES * 4 * 4);
    float* ad1 = (float*)alloc((size_t)LNODES * 4 * 4);
    float* as2 = (float*)alloc((size_t)LNODES * 4 * 4);
    float* ad2 = (float*)alloc((size_t)LNODES * 4 * 4);
    int* row_ptr = (int*)alloc((size_t)(LNODES + 1) * 4);
    int* counts  = (int*)alloc((size_t)LNODES * 4);
    int* cursor  = (int*)alloc((size_t)LNODES * 4);
    int* col     = (int*)alloc((size_t)(NEDGES + LNODES) * 4);
    float* pooled = (float*)alloc(512 * 4);
    float* hfc    = (float*)alloc(512 * 4);

    // ---- CSR-by-destination build (shared by both GAT layers) ----
    csr_count_init<<<LNODES / 256, 256, 0, stream>>>(counts);
    csr_count_edges<<<NEDGES / 256, 256, 0, stream>>>(ei, counts);
    csr_scan<<<1, 1024, 0, stream>>>(counts, row_ptr);
    csr_selfloop<<<LNODES / 256, 256, 0, stream>>>(row_ptr, cursor, col);
    csr_fill<<<NEDGES / 256, 256, 0, stream>>>(ei, cursor, col);

    // ---- pad ragged-K operands once (removes guards from hot GEMM loop) ----
    pad_cols<<<(LNODES * KPAD + 255) / 256, 256, 0, stream>>>(seq, seqp, LNODES, 26, KPAD);
    pad_rows<<<(KPAD * 1024 + 255) / 256, 256, 0, stream>>>(W1, W1p, 26, KPAD, 1024);

    // ---- input MLP: x = relu(seq @ W1) @ W2 ----
    gemm_f32_wmma<<<dim3(1024 / 64, LNODES / 64), 128, 0, stream>>>(
        seqp, W1p, h0, 1024, KPAD, 1);
    gemm_f32_wmma<<<dim3(256 / 64, LNODES / 64), 128, 0, stream>>>(
        h0, W2, xbuf, 256, 1024, 0);

    // ---- GAT layer 1 (H=4, C=64) ----
    gemm_f32_wmma<<<dim3(256 / 64, LNODES / 64), 128, 0, stream>>>(
        xbuf, Wg1, h1, 256, 256, 0);
    attn_scores<<<(LNODES * 4) / 256, 256, 0, stream>>>(h1, asrc1, adst1, as1, ad1, 64);
    gat_aggregate<256><<<LNODES, 256, 0, stream>>>(h1, as1, ad1, row_ptr, col, b1, out1);

    // ---- GAT layer 2 (H=4, C=128) ----
    gemm_f32_wmma<<<dim3(512 / 64, LNODES / 64), 128, 0, stream>>>(
        out1, Wg2, h2, 512, 256, 0);
    attn_scores<<<(LNODES * 4) / 256, 256, 0, stream>>>(h2, asrc2, adst2, as2, ad2, 128);
    gat_aggregate<512><<<LNODES, 256, 0, stream>>>(h2, as2, ad2, row_ptr, col, b2, out2);

    // ---- sum-pool + FC head ----
    zero_f32<<<2, 256, 0, stream>>>(pooled, 512);
    pool_cols512<<<64, 256, 0, stream>>>(out2, pooled);
    fc_kernel<<<2, 256, 0, stream>>>(pooled, fcW, fcb, hfc, 512, 512, 1);
    fc_kernel<<<2, 256, 0, stream>>>(hfc, outW, outb, (float*)d_out, 512, 489, 0);
}